// DS2_3040836845752
// MI455X (gfx1250) — compile-verified
//
#include <hip/hip_runtime.h>
#include <stdint.h>

// ----------------------------------------------------------------------------
// mass_prototype: out[g*8 + k] = in[g] * u[g%32][k],  u = beta^2 / rowsum(beta^2)
// Memory-bound streaming kernel (288 MB traffic, ~12.4 us floor @ 23.3 TB/s).
// CDNA5 path: GLOBAL_LOAD_ASYNC_TO_LDS_B32 4-deep pipeline (ASYNCcnt) for the
// input stream, non-temporal 128-bit stores for the 256 MB output.
// ----------------------------------------------------------------------------

typedef float v4f __attribute__((ext_vector_type(4)));

// Per-lane async copy of one dword global -> LDS. VDST carries the LDS byte
// address (low 32 bits of the flat pointer per ISA sec 10.2 aperture rules).
#define ASYNC_LOAD_B32(lds_u32, gaddr_u64)                                     \
  asm volatile("global_load_async_to_lds_b32 %0, %1, off"                      \
               :: "v"(lds_u32), "v"(gaddr_u64) : "memory")

#define WAIT_ASYNCCNT(n)                                                       \
  asm volatile("s_wait_asynccnt " #n ::: "memory")

#define PIPE_DEPTH 4   // power of two
#define CHUNK 256      // elements per block-iteration (== blockDim.x)

__global__ __launch_bounds__(256) void mass_proto_kernel(
    const float* __restrict__ in, const float* __restrict__ beta,
    float* __restrict__ out, long long n_in) {
  __shared__ float usm[256];                 // beta^2 then u, row-major [32][8]
  __shared__ float stage[PIPE_DEPTH][CHUNK]; // async staging buffers

  const int tid = threadIdx.x;
  const long long nchunks = n_in >> 8;       // full 256-element chunks
  const long long start   = blockIdx.x;
  const long long stride  = gridDim.x;

  // ---- prologue: kick off the async pipeline before anything else ----------
  #pragma unroll
  for (int d = 0; d < PIPE_DEPTH; ++d) {
    long long cc = start + (long long)d * stride;
    if (cc < nchunks) {
      unsigned lds = (unsigned)(uintptr_t)&stage[d][tid];
      unsigned long long ga =
          (unsigned long long)(uintptr_t)(in + cc * CHUNK + tid);
      ASYNC_LOAD_B32(lds, ga);
    }
  }

  // ---- build u table in LDS (overlapped with the in-flight async loads) ----
  float b = beta[tid];            // 256 = D*K entries, one per thread
  usm[tid] = b * b;
  __syncthreads();
  {
    int irow = tid >> 3;          // feature row i of this (i,k) entry
    float s = 0.f;
    #pragma unroll
    for (int kk = 0; kk < 8; ++kk) s += usm[(irow << 3) + kk];
    float uval = (b * b) / s;
    __syncthreads();
    usm[tid] = uval;              // usm[i*8 + k] = u[i][k]
  }
  __syncthreads();

  // cache this lane's u row (i = g & 31 == tid & 31 since chunks are 256-aligned)
  const int i = tid & 31;
  v4f uA, uB;
  #pragma unroll
  for (int kk = 0; kk < 4; ++kk) uA[kk] = usm[i * 8 + kk];
  #pragma unroll
  for (int kk = 0; kk < 4; ++kk) uB[kk] = usm[i * 8 + 4 + kk];

  int buf = 0;
  long long c = start;
  long long cissue = start + (long long)PIPE_DEPTH * stride;

  // ---- main loop: 4 async loads in flight; wait oldest, consume, reissue ---
  for (; cissue < nchunks; c += stride, cissue += stride) {
    WAIT_ASYNCCNT(3);                         // oldest buffer landed (in-order)
    float x = stage[buf][tid];                // ds_load_b32
    long long g = c * CHUNK + tid;
    v4f lo = uA * x;
    v4f hi = uB * x;
    v4f* o = (v4f*)(out + (g << 3));
    __builtin_nontemporal_store(lo, o);       // global_store_b128 (NT)
    __builtin_nontemporal_store(hi, o + 1);
    // reissue into the buffer we just drained (stores above fence the ds read)
    unsigned lds = (unsigned)(uintptr_t)&stage[buf][tid];
    unsigned long long ga =
        (unsigned long long)(uintptr_t)(in + cissue * CHUNK + tid);
    ASYNC_LOAD_B32(lds, ga);
    buf = (buf + 1) & (PIPE_DEPTH - 1);
  }

  // ---- drain remaining in-flight buffers -----------------------------------
  for (; c < nchunks; c += stride) {
    WAIT_ASYNCCNT(0);
    float x = stage[buf][tid];
    long long g = c * CHUNK + tid;
    v4f lo = uA * x;
    v4f hi = uB * x;
    v4f* o = (v4f*)(out + (g << 3));
    __builtin_nontemporal_store(lo, o);
    __builtin_nontemporal_store(hi, o + 1);
    buf = (buf + 1) & (PIPE_DEPTH - 1);
  }

  // ---- tail (n_in not a multiple of 256; not hit for this problem size) ----
  if (blockIdx.x == 0) {
    long long g = nchunks * CHUNK + tid;
    if (g < n_in) {
      float x = in[g];                        // i = g&31 == tid&31 still holds
      v4f lo = uA * x;
      v4f hi = uB * x;
      v4f* o = (v4f*)(out + (g << 3));
      __builtin_nontemporal_store(lo, o);
      __builtin_nontemporal_store(hi, o + 1);
    }
  }
}

extern "C" void kernel_launch(void* const* d_in, const int* in_sizes, int n_in_cnt,
                              void* d_out, int out_size, void* d_ws, size_t ws_size,
                              hipStream_t stream) {
  (void)n_in_cnt; (void)out_size; (void)d_ws; (void)ws_size;
  const float* in   = (const float*)d_in[0];   // (4,256,256,32) f32
  const float* beta = (const float*)d_in[1];   // (32,8) f32
  float* out = (float*)d_out;                  // (4,256,256,32,8) f32
  long long n_in = (long long)in_sizes[0];     // 8,388,608

  // 2048 blocks x 8 wave32 = 16384 waves; 16 chunk-iterations per block.
  mass_proto_kernel<<<2048, 256, 0, stream>>>(in, beta, out, n_in);
}